// GCS_6485400617418
// MI455X (gfx1250) — compile-verified
//
#include <hip/hip_runtime.h>

typedef float v2f __attribute__((ext_vector_type(2)));
typedef float v8f __attribute__((ext_vector_type(8)));

#define T_DIM  64
#define C_DIM  256
#define H_DIM  56
#define W_DIM  56
#define HALF_C 64
#define HW     (H_DIM * W_DIM)   // 3136
#define CHW    (C_DIM * HW)      // 802816
#define PLANE  HW                // one gate plane per t

// ---------------------------------------------------------------------------
// Phase 1: gate = tanh(conv3d_{1 out ch}(x_half) + b) via V_WMMA_F32_16X16X4_F32
//   M (16)  = 16 consecutive input columns w0-1 .. w0+14 of one h-row
//   N (0-2) = kw tap of the 3x3x3 filter (cols 3..15 fed zeros)
//   K (576) = (kd, kh, c) reduction, 4 channels per WMMA step
// Output: 14 gate values per tile via diagonal combine D[i+kw][kw] in LDS.
// Grid: 64 t x 2 sides x 56 h x 4 wtiles = 28672 tiles = 3584 blocks x 8 waves.
// ---------------------------------------------------------------------------
__global__ __launch_bounds__(256)
void gate_conv_wmma(const float* __restrict__ x,
                    const float* __restrict__ w_l, const float* __restrict__ b_l,
                    const float* __restrict__ w_r, const float* __restrict__ b_r,
                    float* __restrict__ gates /* [2][64][3136] */) {
  __shared__ float wlds[HALF_C * 27];  // this block's filter (s is block-uniform)
  __shared__ float dbuf[8][3][16];     // per-wave D columns 0..2

  const int tid   = threadIdx.x;
  const int waveI = tid >> 5;
  const int lane  = tid & 31;
  const int m     = lane & 15;          // A row; B/D column within tile
  const int koff  = (lane >> 4) << 1;   // K sub-offset: lanes 0-15 -> k0/k1, 16-31 -> k2/k3

  // 28 blocks per (t,s): 8 waves/block * 28 = 224 = 56 h-rows * 4 w-tiles
  const int ts  = blockIdx.x / 28;
  const int sub = blockIdx.x % 28;
  const int t   = ts >> 1;
  const int s   = ts & 1;               // 0 = left half, 1 = right half

  const float* wsel = s ? w_r : w_l;
  for (int i = tid; i < HALF_C * 27; i += 256) wlds[i] = wsel[i];
  __syncthreads();

  const int gw = sub * 8 + waveI;       // 0..223
  const int wt = gw & 3;                // w-tile
  const int h  = gw >> 2;               // h row

  const int  w_in = wt * 14 - 1 + m;    // input column this lane's A rows sample
  const bool vw   = (w_in >= 0) && (w_in < W_DIM);

  // B column gating: only kw = 0..2 carry weights
  const int   nn = (m < 3) ? m : 0;
  const float bm = (m < 3) ? 1.0f : 0.0f;

  v8f acc = {};
  #pragma unroll
  for (int kd = 0; kd < 3; ++kd) {
    const int  td = (t + kd + T_DIM - 1) & (T_DIM - 1);     // cyclic T pad
    const long tb = (long)td * CHW + (long)(s * HALF_C) * HW;
    #pragma unroll
    for (int kh = 0; kh < 3; ++kh) {
      const int  hh = h + kh - 1;
      const bool va = vw && (hh >= 0) && (hh < H_DIM);      // zero H/W pad
      const float* xp   = x + tb + hh * W_DIM + w_in;
      const int    wbase = kd * 9 + kh * 3;
      for (int c0 = 0; c0 < HALF_C; c0 += 4) {
        const int c = c0 + koff;
        v2f a, b;
        a.x = va ? xp[(size_t)c * HW]       : 0.0f;
        a.y = va ? xp[(size_t)(c + 1) * HW] : 0.0f;
        b.x = bm * wlds[c * 27 + wbase + nn];
        b.y = bm * wlds[(c + 1) * 27 + wbase + nn];
        acc = __builtin_amdgcn_wmma_f32_16x16x4_f32(
            /*neg_a=*/false, a, /*neg_b=*/false, b,
            /*c_mod=*/(short)0, acc, /*reuse_a=*/false, /*reuse_b=*/false);
      }
    }
  }

  // D layout: lane holds column n = lane&15, rows M = (lane<16 ? 0..7 : 8..15)
  if (m < 3) {
    const int mrow = (lane >> 4) << 3;
    #pragma unroll
    for (int v = 0; v < 8; ++v) dbuf[waveI][m][mrow + v] = acc[v];
  }
  __syncthreads();  // conservative; LDS ops are wave-ordered but keep it safe

  if (lane < 14) {
    float g = dbuf[waveI][0][lane]
            + dbuf[waveI][1][lane + 1]
            + dbuf[waveI][2][lane + 2];
    g += s ? b_r[0] : b_l[0];
    g = tanhf(g);
    gates[(size_t)s * T_DIM * PLANE + (size_t)t * PLANE
          + h * W_DIM + wt * 14 + lane] = g;
  }
}

// ---------------------------------------------------------------------------
// Phase 2: streaming combine (HBM-bound, float4 vectorized)
//   c <  64 : out = Gl[t+1]*x[t+1] + x[t] - Gl[t]*x[t]
//   c < 128 : out = Gr[t-1]*x[t-1] + x[t] - Gr[t]*x[t]
//   else    : out = x[t]
// ---------------------------------------------------------------------------
__global__ __launch_bounds__(256)
void shift_gate_apply(const float* __restrict__ x,
                      const float* __restrict__ gates,
                      float* __restrict__ out) {
  const size_t i4   = (size_t)blockIdx.x * 256 + threadIdx.x;
  const size_t flat = i4 * 4;
  if (flat >= (size_t)T_DIM * CHW) return;

  const int t = (int)(flat / CHW);
  const int r = (int)(flat % CHW);
  const int c = r / HW;
  const int p = r % HW;  // 4-aligned (HW % 4 == 0)

  const float4 xc = *(const float4*)(x + flat);
  float4 o;
  if (c < 2 * HALF_C) {
    const int s  = c >> 6;
    const int tn = s ? ((t + T_DIM - 1) & (T_DIM - 1)) : ((t + 1) & (T_DIM - 1));
    const float4 xn = *(const float4*)(x + (size_t)tn * CHW + (size_t)c * HW + p);
    const float* G  = gates + (size_t)s * T_DIM * PLANE;
    const float4 gc = *(const float4*)(G + (size_t)t  * PLANE + p);
    const float4 gn = *(const float4*)(G + (size_t)tn * PLANE + p);
    o.x = gn.x * xn.x + xc.x - gc.x * xc.x;
    o.y = gn.y * xn.y + xc.y - gc.y * xc.y;
    o.z = gn.z * xn.z + xc.z - gc.z * xc.z;
    o.w = gn.w * xn.w + xc.w - gc.w * xc.w;
  } else {
    o = xc;
  }
  *(float4*)(out + flat) = o;
}

extern "C" void kernel_launch(void* const* d_in, const int* in_sizes, int n_in,
                              void* d_out, int out_size, void* d_ws, size_t ws_size,
                              hipStream_t stream) {
  const float* x   = (const float*)d_in[0];
  const float* w_l = (const float*)d_in[1];
  const float* b_l = (const float*)d_in[2];
  const float* w_r = (const float*)d_in[3];
  const float* b_r = (const float*)d_in[4];
  float* out   = (float*)d_out;
  float* gates = (float*)d_ws;  // needs 2*64*3136*4 = 1,605,632 bytes

  // Phase 1: 28672 tiles, one wave each -> 3584 blocks of 8 waves
  gate_conv_wmma<<<3584, 256, 0, stream>>>(x, w_l, b_l, w_r, b_r, gates);

  // Phase 2: 51,380,224 elems / 4 per thread / 256 per block = 50,176 blocks
  shift_gate_apply<<<50176, 256, 0, stream>>>(x, gates, out);
}